// GraphConvolution_8297876816010
// MI455X (gfx1250) — compile-verified
//
#include <hip/hip_runtime.h>

typedef float v2f __attribute__((ext_vector_type(2)));
typedef float v8f __attribute__((ext_vector_type(8)));

#define IN_C  256
#define OUT_C 128
#define EPB   512   // edges per block in SpMM stage

// ---------------------------------------------------------------------------
// Zero-init output (harness poisons d_out with 0xAA before timing).
// ---------------------------------------------------------------------------
__global__ void gcn_zero_kernel(float* __restrict__ out, int total) {
    int i = blockIdx.x * blockDim.x + threadIdx.x;
    if (i < total) out[i] = 0.0f;
}

// ---------------------------------------------------------------------------
// H = X @ W^T  via V_WMMA_F32_16X16X4_F32, software-pipelined.
// One wave computes a 16-row stripe across all 128 output channels
// (8 accumulator tiles of 16x16). K loop steps by 4, prefetching the next
// k-step's A fragment + 8 B fragments into distinct registers while the
// current 8 WMMAs execute (one amortized s_wait per 8 WMMAs).
//
// A fragment (16x4 f32, MxK):  lane<16 : v0=A[M=lid][k+0], v1=A[M=lid][k+1]
//                              lane>=16: v0=A[M=lid][k+2], v1=A[M=lid][k+3]
// B fragment (4x16 f32, KxN):  lane<16 : v0=B[k+0][N=lid], v1=B[k+1][N=lid]
//                              lane>=16: v0=B[k+2][N=lid], v1=B[k+3][N=lid]
// with B[k][o] = W[o][k]  ->  both fragments are contiguous float2 loads.
//
// C/D (16x16 f32): VGPR g, lanes 0-15 -> (M=g,  N=lid)
//                          lanes16-31 -> (M=g+8,N=lid)
// ---------------------------------------------------------------------------
__global__ void gcn_gemm_wmma_kernel(const float* __restrict__ X,
                                     const float* __restrict__ W,
                                     float* __restrict__ H,
                                     int N) {
    const int wavesPerBlock = blockDim.x >> 5;
    const int wave = blockIdx.x * wavesPerBlock + (threadIdx.x >> 5);
    const int lane = threadIdx.x & 31;
    const int m0 = wave * 16;
    if (m0 >= N) return;

    const int half = lane >> 4;   // 0: K pair {0,1}, 1: K pair {2,3}
    const int lid  = lane & 15;

    // Row this lane feeds into the A fragment (clamp for ragged tail).
    int arow = m0 + lid;
    if (arow >= N) arow = N - 1;
    const float* __restrict__ xrow  = X + (size_t)arow * IN_C + 2 * half;
    const float* __restrict__ wbase = W + (size_t)lid  * IN_C + 2 * half;

    v8f acc[8] = {};   // 8 output tiles of 16x16 -> columns 0..127

    // ---- software pipeline: prologue loads for k = 0 ----
    v2f a_cur = *(const v2f*)(xrow);
    v2f b_cur[8];
#pragma unroll
    for (int t = 0; t < 8; ++t)
        b_cur[t] = *(const v2f*)(wbase + (size_t)t * 16 * IN_C);

#pragma unroll 2
    for (int k = 0; k < IN_C; k += 4) {
        // Prefetch next k-step into *distinct* registers (clamped at tail:
        // reloading the same fragment is harmless and branch-free).
        const int kn = (k + 4 < IN_C) ? (k + 4) : k;
        v2f a_nxt = *(const v2f*)(xrow + kn);
        v2f b_nxt[8];
#pragma unroll
        for (int t = 0; t < 8; ++t)
            b_nxt[t] = *(const v2f*)(wbase + (size_t)t * 16 * IN_C + kn);

        // 8-WMMA burst on the fragments loaded one iteration ago.
#pragma unroll
        for (int t = 0; t < 8; ++t)
            acc[t] = __builtin_amdgcn_wmma_f32_16x16x4_f32(
                /*neg_a=*/false, a_cur, /*neg_b=*/false, b_cur[t],
                /*c_mod=*/(short)0, acc[t],
                /*reuse_a=*/false, /*reuse_b=*/false);

        a_cur = a_nxt;
#pragma unroll
        for (int t = 0; t < 8; ++t) b_cur[t] = b_nxt[t];
    }

    // Store 16x128 stripe of H.
    // Fast path (full tile): 64 unguarded stores from one base address with
    // immediate offsets -> no per-store exec-mask churn.
    float* __restrict__ hbase = H + (size_t)(m0 + 8 * half) * OUT_C + lid;
    if (m0 + 16 <= N) {
#pragma unroll
        for (int t = 0; t < 8; ++t) {
            float* __restrict__ hp = hbase + t * 16;
#pragma unroll
            for (int g = 0; g < 8; ++g)
                hp[(size_t)g * OUT_C] = acc[t][g];
        }
    } else {
        // Ragged tail tile: per-row guard.
#pragma unroll
        for (int t = 0; t < 8; ++t) {
            const int col = t * 16 + lid;
#pragma unroll
            for (int g = 0; g < 8; ++g) {
                const int row = m0 + g + 8 * half;
                if (row < N) H[(size_t)row * OUT_C + col] = acc[t][g];
            }
        }
    }
}

// ---------------------------------------------------------------------------
// out[r] += v * H[c]  for each edge (r, c, v); rows sorted ascending.
// Block = 128 threads, thread owns one output channel. 512 edges staged in
// LDS per block; register accumulation while the row is unchanged, atomic
// flush on row change (handles cross-block row splits too).
// ---------------------------------------------------------------------------
__global__ void gcn_spmm_kernel(const int* __restrict__ A_rows,
                                const int* __restrict__ A_cols,
                                const float* __restrict__ A_vals,
                                const float* __restrict__ H,
                                float* __restrict__ out,
                                int E) {
    __shared__ int   s_row[EPB];
    __shared__ int   s_col[EPB];
    __shared__ float s_val[EPB];

    const int e0 = blockIdx.x * EPB;
    int n = E - e0;
    if (n > EPB) n = EPB;
    if (n <= 0) return;

    for (int i = threadIdx.x; i < n; i += blockDim.x) {
        s_row[i] = A_rows[e0 + i];
        s_col[i] = A_cols[e0 + i];
        s_val[i] = A_vals[e0 + i];
    }
    __syncthreads();

    const int c = threadIdx.x;       // channel 0..127
    float acc = 0.0f;
    int cur = s_row[0];

    for (int i = 0; i < n; ++i) {
        const int r = s_row[i];
        if (r != cur) {
            atomicAdd(&out[(size_t)cur * OUT_C + c], acc);
            acc = 0.0f;
            cur = r;
        }
        acc += s_val[i] * H[(size_t)s_col[i] * OUT_C + c];
    }
    atomicAdd(&out[(size_t)cur * OUT_C + c], acc);
}

// ---------------------------------------------------------------------------
extern "C" void kernel_launch(void* const* d_in, const int* in_sizes, int n_in,
                              void* d_out, int out_size, void* d_ws, size_t ws_size,
                              hipStream_t stream) {
    const float* X      = (const float*)d_in[0];
    const float* W      = (const float*)d_in[1];
    const int*   A_rows = (const int*)d_in[2];
    const int*   A_cols = (const int*)d_in[3];
    const float* A_vals = (const float*)d_in[4];
    float* out = (float*)d_out;
    float* H   = (float*)d_ws;        // N * OUT_C floats of scratch

    const int N = in_sizes[0] / IN_C;   // 100000
    const int E = in_sizes[2];          // 1600000

    // 1) zero the output accumulator
    gcn_zero_kernel<<<(out_size + 255) / 256, 256, 0, stream>>>(out, out_size);

    // 2) dense GEMM via fp32 WMMA: 8 waves per block, one 16-row stripe each
    const int tiles  = (N + 15) / 16;
    const int blocks = (tiles + 7) / 8;
    gcn_gemm_wmma_kernel<<<blocks, 256, 0, stream>>>(X, W, H, N);

    // 3) sorted-row sparse scatter-accumulate
    gcn_spmm_kernel<<<(E + EPB - 1) / EPB, 128, 0, stream>>>(A_rows, A_cols, A_vals, H, out, E);
}